// ChamferLoss_25074019074108
// MI455X (gfx1250) — compile-verified
//
#include <hip/hip_runtime.h>

typedef __attribute__((ext_vector_type(2))) float v2f;
typedef __attribute__((ext_vector_type(8))) float v8f;

// Zero the per-batch accumulators in workspace (harness poisons ws with 0xAA).
__global__ void chamfer_init(float* __restrict__ acc, int n) {
    int i = blockIdx.x * blockDim.x + threadIdx.x;
    if (i < n) acc[i] = 0.0f;
}

// One wave owns one (batch, 16-row strip of P) pair and sweeps all columns of Q.
// Distance tile via V_WMMA_F32_16X16X4_F32:
//   A row n = (-2Px, -2Py, -2Pz, |P|^2),  B col m = (Qx, Qy, Qz, 1),  C[.][m] = |Q|^2
//   D = |P|^2 - 2 P.Q + |Q|^2
// Accumulates sum over the strip's 16 rows of min_m max(0, D[n][m]) into acc[b].
__global__ __launch_bounds__(256)
void chamfer_rowmin_pass(const float* __restrict__ P,   // (B, nP, 3) -> rows
                         const float* __restrict__ Q,   // (B, nQ, 3) -> cols
                         float* __restrict__ acc,       // (B,) partial sums
                         int nP, int nQ, int B)
{
    const int lane   = threadIdx.x & 31;
    const int wid    = blockIdx.x * (blockDim.x >> 5) + (threadIdx.x >> 5);
    const int strips = nP >> 4;
    const int nWaves = B * strips;
    if (wid >= nWaves) return;                  // whole-wave exit: EXEC stays all-1s

    const int b     = wid / strips;
    const int strip = wid - b * strips;

    const float* __restrict__ Pb = P + (size_t)b * nP * 3;
    const float* __restrict__ Qb = Q + (size_t)b * nQ * 3;

    const bool hi = (lane >= 16);
    const int  r  = (strip << 4) + (lane & 15);

    // Build loop-invariant A operand (16x4 f32, 2 VGPRs).
    // VGPR0: lanes0-15 K=0, lanes16-31 K=2 ; VGPR1: lanes0-15 K=1, lanes16-31 K=3
    const float px = Pb[3 * r + 0];
    const float py = Pb[3 * r + 1];
    const float pz = Pb[3 * r + 2];
    const float pn = px * px + py * py + pz * pz;
    v2f a;
    a.x = hi ? (-2.0f * pz) : (-2.0f * px);
    a.y = hi ? pn           : (-2.0f * py);

    const float INF = __builtin_inff();
    v8f mins = { INF, INF, INF, INF, INF, INF, INF, INF };

    const int tiles = nQ >> 4;
    const float* __restrict__ qptr = Qb + 3 * (lane & 15);

    #pragma unroll 4
    for (int t = 0; t < tiles; ++t) {
        const float qx = qptr[0];
        const float qy = qptr[1];
        const float qz = qptr[2];
        qptr += 48;                             // advance 16 points * 3 floats
        const float qn = qx * qx + qy * qy + qz * qz;

        // B operand (4x16 f32): VGPR0 lanes0-15 K=0 (Qx), lanes16-31 K=2 (Qz)
        //                       VGPR1 lanes0-15 K=1 (Qy), lanes16-31 K=3 (1.0)
        v2f bv;
        bv.x = hi ? qz   : qx;
        bv.y = hi ? 1.0f : qy;

        // C: |Q|^2 broadcast down each column (same value in all 8 C VGPRs per lane)
        v8f c = { qn, qn, qn, qn, qn, qn, qn, qn };

        v8f d = __builtin_amdgcn_wmma_f32_16x16x4_f32(
            /*neg_a=*/false, a, /*neg_b=*/false, bv,
            /*c_mod=*/(short)0, c, /*reuse_a=*/false, /*reuse_b=*/false);

        #pragma unroll
        for (int j = 0; j < 8; ++j)
            mins[j] = fminf(mins[j], d[j]);
    }

    // Clamp (max(0,.) commutes with min), reduce each row-min across its
    // 16-lane half, sum the 16 row minima of this strip, accumulate.
    float s = 0.0f;
    #pragma unroll
    for (int j = 0; j < 8; ++j) {
        float v = fmaxf(mins[j], 0.0f);
        v = fminf(v, __shfl_xor(v, 1, 32));
        v = fminf(v, __shfl_xor(v, 2, 32));
        v = fminf(v, __shfl_xor(v, 4, 32));
        v = fminf(v, __shfl_xor(v, 8, 32));     // all lanes in half hold row min
        s += v;                                  // lane<16: rows 0-7 ; lane>=16: rows 8-15
    }
    s += __shfl_xor(s, 16, 32);                  // combine halves: sum of 16 row mins
    if (lane == 0) atomicAdd(&acc[b], s);
}

// loss = mean_b( w[b] * (accX[b]/N + accY[b]/M) )
__global__ void chamfer_finalize(const float* __restrict__ acc,
                                 const float* __restrict__ w,
                                 float* __restrict__ out,
                                 int B, float invN, float invM)
{
    if (threadIdx.x == 0 && blockIdx.x == 0) {
        float s = 0.0f;
        for (int b = 0; b < B; ++b)
            s += w[b] * (acc[b] * invN + acc[B + b] * invM);
        out[0] = s / (float)B;
    }
}

extern "C" void kernel_launch(void* const* d_in, const int* in_sizes, int n_in,
                              void* d_out, int out_size, void* d_ws, size_t ws_size,
                              hipStream_t stream)
{
    const float* x = (const float*)d_in[0];   // (B, N, 3) f32
    const float* y = (const float*)d_in[1];   // (B, M, 3) f32
    const float* w = (const float*)d_in[2];   // (B,)     f32

    const int B = in_sizes[2];
    const int N = in_sizes[0] / (3 * B);
    const int M = in_sizes[1] / (3 * B);

    float* acc = (float*)d_ws;                // 2*B floats: [0,B)=x-side, [B,2B)=y-side

    chamfer_init<<<1, 64, 0, stream>>>(acc, 2 * B);

    // Pass 1: rows = x (strip mins over y)  -> acc[0..B)
    {
        int waves  = B * (N >> 4);
        int blocks = (waves * 32 + 255) / 256;
        chamfer_rowmin_pass<<<blocks, 256, 0, stream>>>(x, y, acc, N, M, B);
    }
    // Pass 2: rows = y (strip mins over x)  -> acc[B..2B)
    {
        int waves  = B * (M >> 4);
        int blocks = (waves * 32 + 255) / 256;
        chamfer_rowmin_pass<<<blocks, 256, 0, stream>>>(y, x, acc + B, M, N, B);
    }

    chamfer_finalize<<<1, 32, 0, stream>>>(acc, w, (float*)d_out, B,
                                           1.0f / (float)N, 1.0f / (float)M);
}